// KeypointMatchingModel_89988154786376
// MI455X (gfx1250) — compile-verified
//
#include <hip/hip_runtime.h>
#include <hip/hip_bf16.h>
#include <stdint.h>

// ---------------------------------------------------------------------------
// KeypointMatchingModel for MI455X (gfx1250, wave32).
// - All GEMMs on V_WMMA_F32_16X16X4_F32 (reference is fp32; ~57 GFLOP, ~25MB
//   working set << 192MB L2 -> compute/latency bound, full precision is right).
// - Main (NN) GEMM stages A/B panels into LDS with the Tensor Data Mover
//   (tensor_load_to_lds + s_wait_tensorcnt), double buffered, with TDM
//   hardware padding giving a 36-float LDS row stride (bank-conflict free
//   ds_load_b64 fragment reads).
// ---------------------------------------------------------------------------

typedef __attribute__((ext_vector_type(2))) float v2f;
typedef __attribute__((ext_vector_type(8))) float v8f;
typedef __attribute__((ext_vector_type(4))) unsigned int v4u;
typedef __attribute__((ext_vector_type(8))) int v8i;
typedef __attribute__((ext_vector_type(4))) int v4i;

#define NN     1024   // B*N nodes per side
#define NB     8      // batch
#define NPER   128    // nodes per graph
#define CIN    512
#define COUT   512
#define HM     128    // MLP hidden
#define NE     16384  // edges per side
#define NSTEPS 10

#if defined(__has_builtin)
#if __has_builtin(__builtin_amdgcn_tensor_load_to_lds) && \
    __has_builtin(__builtin_amdgcn_s_wait_tensorcnt)
#define USE_TDM 1
#endif
#endif
#ifndef USE_TDM
#define USE_TDM 0
#endif

// ---------------------------- small utility kernels ------------------------

__global__ void k_fill0(float* p, int n) {
  int i = blockIdx.x * blockDim.x + threadIdx.x;
  if (i < n) p[i] = 0.f;
}

__global__ void k_count2(const int* __restrict__ src, const int* __restrict__ dst,
                         float* cnt_src, float* cnt_dst, int e_cnt) {
  int e = blockIdx.x * blockDim.x + threadIdx.x;
  if (e < e_cnt) {
    atomicAdd(&cnt_src[src[e]], 1.f);
    atomicAdd(&cnt_dst[dst[e]], 1.f);
  }
}

__global__ void k_invert(float* p, int n) {
  int i = blockIdx.x * blockDim.x + threadIdx.x;
  if (i < n) p[i] = 1.f / fmaxf(p[i], 1.f);
}

// ------------------------- TDM 2-D tile -> LDS issue ------------------------
// Builds the Tensor DMA Descriptor (D#) per CDNA5 ISA 8.3/8.4 and issues
// TENSOR_LOAD_TO_LDS (6-arg clang-23 form). data_size=4B; tile rows (tile_d0
// dwords, contiguous in memory) are stored to LDS with a hardware pad of 4
// dwords every 32 dwords -> LDS row stride of 36 floats when tile_d0 == 32.
#if USE_TDM
__device__ __forceinline__ void tdm_load_2d_f32(unsigned lds_off, const float* gptr,
                                                unsigned tile_d0, unsigned tile_d1,
                                                unsigned stride0 /*elements*/) {
  unsigned long long ga = (unsigned long long)(uintptr_t)gptr;
  v4u g0;
  g0[0] = 1u;                                    // count=1 valid descriptor
  g0[1] = lds_off;                               // D#.lds_addr (bytes)
  g0[2] = (unsigned)(ga & 0xFFFFFFFFull);        // global_addr[31:0]
  g0[3] = (unsigned)((ga >> 32) & 0x01FFFFFFull) // global_addr[56:32]
          | 0x80000000u;                         // type = 2 ("image")
  v8i g1;
  // wg_mask=0 | data_size(4B)=2<<16 | pad_en=1<<20 | pad_interval(32dw)=4<<22
  // | pad_amount(4dw)=3<<25
  g1[0] = (int)((2u << 16) | (1u << 20) | (4u << 22) | (3u << 25));
  g1[1] = (int)((tile_d0 & 0xFFFFu) << 16);      // tensor_dim0[15:0] (== tile)
  g1[2] = (int)(((tile_d0 >> 16) & 0xFFFFu) | ((tile_d1 & 0xFFFFu) << 16));
  g1[3] = (int)(((tile_d1 >> 16) & 0xFFFFu) | ((tile_d0 & 0xFFFFu) << 16));
  g1[4] = (int)(tile_d1 & 0xFFFFu);              // tile_dim1 | tile_dim2=0
  g1[5] = (int)stride0;                          // tensor_dim0_stride[31:0]
  g1[6] = 0;                                     // stride0 hi | stride1 lo
  g1[7] = 0;
  v4i z4 = {0, 0, 0, 0};
  v8i z8 = {0, 0, 0, 0, 0, 0, 0, 0};
  __builtin_amdgcn_tensor_load_to_lds(g0, g1, z4, z4, z8, 0);
}
#endif

// ------------------------------ WMMA f32 GEMM (NN) --------------------------
// C[M,N] (=/+=) A[M,K] @ B[K,N] (+bias). Block: 256 thr = 8 waves tiling
// 64(M) x 32(N); K consumed in 32-wide chunks staged in LDS:
//   Asm: [m][k] 64x(32+4pad), Bsm: [k][n] 32x(32+4pad)
// Fragment layouts per ISA 7.12.2 (A/B: half-waves hold K 0,1 / 2,3).
// All dims used here divide the tiles (M%64==0, N%32==0, K%32==0).

#define LDSP 36  // padded LDS row stride (floats) == TDM pad result

__global__ __launch_bounds__(256)
void k_gemm_nn(const float* __restrict__ A, const float* __restrict__ B,
               const float* __restrict__ bias, float* __restrict__ C,
               int K, int lda, int ldb, int ldc, int accumulate) {
  __shared__ float Asm[2][64][LDSP];
  __shared__ float Bsm[2][32][LDSP];
  const int tid  = threadIdx.x;
  const int w    = tid >> 5;
  const int lane = tid & 31;
  const int mw   = (w & 3) * 16;        // wave's M sub-tile in block
  const int nw   = (w >> 2) * 16;       // wave's N sub-tile in block
  const int m0   = blockIdx.y * 64;
  const int n0   = blockIdx.x * 32;
  const int r16  = lane & 15;
  const int kh   = (lane >> 4) * 2;
  const int NC   = K >> 5;

  v8f acc = {0.f, 0.f, 0.f, 0.f, 0.f, 0.f, 0.f, 0.f};

#if USE_TDM
  if (w == 0) {  // wave 0 drives the DMA; EXEC-independent per-wave op
    tdm_load_2d_f32((unsigned)(uintptr_t)&Asm[0][0][0],
                    A + (size_t)m0 * lda, 32u, 64u, (unsigned)lda);
    tdm_load_2d_f32((unsigned)(uintptr_t)&Bsm[0][0][0],
                    B + n0, 32u, 32u, (unsigned)ldb);
  }
#endif

  for (int c = 0; c < NC; ++c) {
    const int cur = c & 1;
#if USE_TDM
    if (w == 0) {
      if (c + 1 < NC) {  // prefetch next chunk into other buffer
        const int k1 = (c + 1) << 5;
        tdm_load_2d_f32((unsigned)(uintptr_t)&Asm[cur ^ 1][0][0],
                        A + (size_t)m0 * lda + k1, 32u, 64u, (unsigned)lda);
        tdm_load_2d_f32((unsigned)(uintptr_t)&Bsm[cur ^ 1][0][0],
                        B + (size_t)k1 * ldb + n0, 32u, 32u, (unsigned)ldb);
        __builtin_amdgcn_s_wait_tensorcnt(2);  // chunk c complete (in-order)
      } else {
        __builtin_amdgcn_s_wait_tensorcnt(0);
      }
    }
    __syncthreads();                           // publish chunk c to all waves
#else
    __syncthreads();                           // previous compute done
    {
      const int k0 = c << 5;
      // A panel: 64 rows x 32 floats; 4 threads/row, 2 float4 each
      const int ra = tid >> 2, ca = (tid & 3) * 8;
      const float* ap = A + (size_t)(m0 + ra) * lda + k0 + ca;
      *(float4*)&Asm[cur][ra][ca]     = *(const float4*)(ap);
      *(float4*)&Asm[cur][ra][ca + 4] = *(const float4*)(ap + 4);
      // B panel: 32 rows x 32 floats; 8 threads/row, 1 float4 each
      const int rb = tid >> 3, cb = (tid & 7) * 4;
      *(float4*)&Bsm[cur][rb][cb] =
          *(const float4*)(B + (size_t)(k0 + rb) * ldb + n0 + cb);
    }
    __syncthreads();
#endif
#pragma unroll
    for (int kk = 0; kk < 32; kk += 4) {
      v2f a, b;
      a.x = Asm[cur][mw + r16][kk + kh];       // ds_load_b64, conflict-free
      a.y = Asm[cur][mw + r16][kk + kh + 1];
      b.x = Bsm[cur][kk + kh][nw + r16];
      b.y = Bsm[cur][kk + kh + 1][nw + r16];
      acc = __builtin_amdgcn_wmma_f32_16x16x4_f32(false, a, false, b,
                                                  (short)0, acc, false, false);
    }
    __syncthreads();                           // protect buffers for next DMA
  }

  const int crow = m0 + mw + (lane >> 4) * 8;
  const int ccol = n0 + nw + r16;
  const float bv = bias ? bias[ccol] : 0.f;
#pragma unroll
  for (int r = 0; r < 8; ++r) {
    float* cp = C + (size_t)(crow + r) * ldc + ccol;
    float v = acc[r] + bv;
    *cp = accumulate ? (*cp + v) : v;
  }
}

// --------------- WMMA f32 GEMM, transposed variants (small ops) -------------
// Direct-global fragment loads; used for S_hat (NT, batched) and S^T@R (TN).

template<bool TA, bool TB>
__global__ __launch_bounds__(256)
void k_gemm_wmma(const float* __restrict__ A, const float* __restrict__ B,
                 float* __restrict__ C, int K, int lda, int ldb, int ldc,
                 long long sA, long long sB, long long sC) {
  A += blockIdx.z * sA;
  B += blockIdx.z * sB;
  C += blockIdx.z * sC;
  const int w    = threadIdx.x >> 5;
  const int lane = threadIdx.x & 31;
  const int m0   = blockIdx.y * 64 + (w & 3) * 16;
  const int n0   = blockIdx.x * 32 + (w >> 2) * 16;
  const int r16  = lane & 15;
  const int kh   = (lane >> 4) * 2;

  v8f acc = {0.f, 0.f, 0.f, 0.f, 0.f, 0.f, 0.f, 0.f};
  for (int k0 = 0; k0 < K; k0 += 4) {
    v2f a, b;
    if (!TA) {
      const float* ap = A + (m0 + r16) * lda + (k0 + kh);
      a.x = ap[0];
      a.y = ap[1];
    } else {  // A(m,k) = A[k*lda + m]
      a.x = A[(k0 + kh)     * lda + (m0 + r16)];
      a.y = A[(k0 + kh + 1) * lda + (m0 + r16)];
    }
    if (!TB) {
      b.x = B[(k0 + kh)     * ldb + (n0 + r16)];
      b.y = B[(k0 + kh + 1) * ldb + (n0 + r16)];
    } else {  // B(k,n) = B[n*ldb + k]
      const float* bp = B + (n0 + r16) * ldb + (k0 + kh);
      b.x = bp[0];
      b.y = bp[1];
    }
    acc = __builtin_amdgcn_wmma_f32_16x16x4_f32(false, a, false, b,
                                                (short)0, acc, false, false);
  }
  const int crow = m0 + (lane >> 4) * 8;
  const int ccol = n0 + r16;
#pragma unroll
  for (int r = 0; r < 8; ++r) C[(crow + r) * ldc + ccol] = acc[r];
}

// ------------------------- graph scatter (mean aggr) ------------------------

__global__ void k_scatter_mean(const float* __restrict__ Y, const int* __restrict__ from_i,
                               const int* __restrict__ to_i, const float* __restrict__ invdeg,
                               float* __restrict__ H) {
  const int e  = blockIdx.x;
  const int c4 = threadIdx.x;                    // 128 threads * 4 floats = 512
  const int s  = from_i[e];
  const int d  = to_i[e];
  const float wgt = invdeg[d];
  const float4 v = ((const float4*)(Y + s * CIN))[c4];
  float* o = H + d * CIN + c4 * 4;
  atomicAdd(o + 0, v.x * wgt);
  atomicAdd(o + 1, v.y * wgt);
  atomicAdd(o + 2, v.z * wgt);
  atomicAdd(o + 3, v.w * wgt);
}

// -------------------------- fused relu + LayerNorm --------------------------

__global__ __launch_bounds__(256)
void k_relu_layernorm(const float* __restrict__ H, const float* __restrict__ g,
                      const float* __restrict__ b, float* __restrict__ O, int rows) {
  const int wid  = (blockIdx.x * blockDim.x + threadIdx.x) >> 5;
  const int lane = threadIdx.x & 31;
  if (wid >= rows) return;
  const float* hr = H + wid * COUT;
  float v[16];
  float s1 = 0.f, s2 = 0.f;
#pragma unroll
  for (int k = 0; k < 16; ++k) {
    float x = fmaxf(hr[lane + 32 * k], 0.f);
    v[k] = x;
    s1 += x;
    s2 += x * x;
  }
#pragma unroll
  for (int off = 16; off > 0; off >>= 1) {
    s1 += __shfl_xor(s1, off, 32);
    s2 += __shfl_xor(s2, off, 32);
  }
  const float mu  = s1 * (1.f / (float)COUT);
  const float var = s2 * (1.f / (float)COUT) - mu * mu;
  const float inv = rsqrtf(var + 1e-5f);
  float* orow = O + wid * COUT;
#pragma unroll
  for (int k = 0; k < 16; ++k) {
    const int c = lane + 32 * k;
    orow[c] = (v[k] - mu) * inv * g[c] + b[c];
  }
}

// ------------------------------ row softmax (128) ---------------------------

__global__ __launch_bounds__(256)
void k_softmax128(const float* __restrict__ X, float* __restrict__ O, int rows) {
  const int wid  = (blockIdx.x * blockDim.x + threadIdx.x) >> 5;
  const int lane = threadIdx.x & 31;
  if (wid >= rows) return;
  const float* xr = X + wid * NPER;
  float v[4];
  float m = -3.0e38f;
#pragma unroll
  for (int k = 0; k < 4; ++k) {
    v[k] = xr[lane + 32 * k];
    m = fmaxf(m, v[k]);
  }
#pragma unroll
  for (int off = 16; off > 0; off >>= 1) m = fmaxf(m, __shfl_xor(m, off, 32));
  float s = 0.f;
#pragma unroll
  for (int k = 0; k < 4; ++k) {
    v[k] = __expf(v[k] - m);
    s += v[k];
  }
#pragma unroll
  for (int off = 16; off > 0; off >>= 1) s += __shfl_xor(s, off, 32);
  const float inv = 1.f / s;
  float* orow = O + wid * NPER;
#pragma unroll
  for (int k = 0; k < 4; ++k) orow[lane + 32 * k] = v[k] * inv;
}

// -------------------- counter-based RNG (threefry-style) --------------------

__device__ inline void threefry2x32(unsigned k0, unsigned k1, unsigned c0, unsigned c1,
                                    unsigned& o0, unsigned& o1) {
  const unsigned R[8] = {13u, 15u, 26u, 6u, 17u, 29u, 16u, 24u};
  unsigned ks[3] = {k0, k1, k0 ^ k1 ^ 0x1BD11BDAu};
  unsigned x0 = c0 + ks[0], x1 = c1 + ks[1];
#pragma unroll
  for (int r = 0; r < 20; ++r) {
    x0 += x1;
    const unsigned rot = R[r & 7];
    x1 = (x1 << rot) | (x1 >> (32 - rot));
    x1 ^= x0;
    if ((r & 3) == 3) {
      const int j = (r >> 2) + 1;
      x0 += ks[j % 3];
      x1 += ks[(j + 1) % 3] + (unsigned)j;
    }
  }
  o0 = x0;
  o1 = x1;
}

__global__ void k_rng_normal(float* __restrict__ out, int n, unsigned step) {
  const int i = blockIdx.x * blockDim.x + threadIdx.x;  // 2 values per thread
  if (2 * i >= n) return;
  unsigned r0, r1;
  threefry2x32(42u + step * 0x9E3779B9u, 0x85EBCA6Bu ^ step,
               (unsigned)i, 0xC2B2AE35u, r0, r1);
  const float u1 = (r0 >> 8) * (1.f / 16777216.f);
  const float u2 = (r1 >> 8) * (1.f / 16777216.f);
  const float rad = sqrtf(-2.f * __logf(u1 + 1.1754944e-38f));
  out[2 * i] = rad * __cosf(6.28318530718f * u2);
  if (2 * i + 1 < n) out[2 * i + 1] = rad * __sinf(6.28318530718f * u2);
}

// -------------- pairwise relu(A_i - B_j) . w2 update of S_hat ---------------

__global__ __launch_bounds__(256)
void k_pairwise(const float* __restrict__ Ab, const float* __restrict__ Bb,
                const float* __restrict__ w2, const float* __restrict__ b2,
                float* __restrict__ Shat) {
  __shared__ float At[16][HM + 4];
  __shared__ float Bt[16][HM + 4];
  __shared__ float Ws[HM];
  const int b  = blockIdx.z;
  const int i0 = blockIdx.y * 16;
  const int j0 = blockIdx.x * 16;
  const int tid = threadIdx.x;
  for (int idx = tid; idx < 16 * HM; idx += 256) {
    const int r = idx >> 7, c = idx & (HM - 1);
    At[r][c] = Ab[(b * NPER + i0 + r) * HM + c];
    Bt[r][c] = Bb[(b * NPER + j0 + r) * HM + c];
  }
  if (tid < HM) Ws[tid] = w2[tid];
  __syncthreads();
  const int ti = tid >> 4, tj = tid & 15;
  float acc = b2[0];
#pragma unroll 4
  for (int h = 0; h < HM; ++h) {
    const float d = At[ti][h] - Bt[tj][h];
    acc = fmaf(fmaxf(d, 0.f), Ws[h], acc);
  }
  Shat[(b * NPER + i0 + ti) * NPER + (j0 + tj)] += acc;
}

// ------------------------------- host helpers -------------------------------

struct GnnW {
  const float *W1, *W2, *Wr, *rb, *g, *b, *Fw, *Fb;
};

static void run_gemm(hipStream_t st, const float* A, const float* B, const float* bias,
                     float* C, int M, int N, int K, int lda, int ldb, int ldc,
                     int accumulate) {
  dim3 grid(N / 32, M / 64, 1);
  k_gemm_nn<<<grid, 256, 0, st>>>(A, B, bias, C, K, lda, ldb, ldc, accumulate);
}

// One RelConv + relu/LayerNorm + concat-linear GNN pass over all NN nodes.
// Concat-linear decomposed: OUT = X @ Fw[0:CIN] + Hn @ Fw[CIN:] + Fb.
static void run_gnn(hipStream_t st, const float* X, const int* src, const int* dst,
                    const float* inv_dst, const float* inv_src, const GnnW& w,
                    float* OUT, float* Y1, float* Y2, float* Hb, float* Hn) {
  run_gemm(st, X, w.W1, nullptr, Y1, NN, COUT, CIN, CIN, COUT, COUT, 0);
  run_gemm(st, X, w.W2, nullptr, Y2, NN, COUT, CIN, CIN, COUT, COUT, 0);
  run_gemm(st, X, w.Wr, w.rb,   Hb, NN, COUT, CIN, CIN, COUT, COUT, 0);
  k_scatter_mean<<<NE, 128, 0, st>>>(Y1, src, dst, inv_dst, Hb);  // src -> dst
  k_scatter_mean<<<NE, 128, 0, st>>>(Y2, dst, src, inv_src, Hb);  // dst -> src
  k_relu_layernorm<<<(NN * 32) / 256, 256, 0, st>>>(Hb, w.g, w.b, Hn, NN);
  run_gemm(st, X,  w.Fw,              w.Fb,   OUT, NN, COUT, CIN,  CIN,  COUT, COUT, 0);
  run_gemm(st, Hn, w.Fw + CIN * COUT, nullptr, OUT, NN, COUT, COUT, COUT, COUT, COUT, 1);
}

// --------------------------------- launcher ---------------------------------

extern "C" void kernel_launch(void* const* d_in, const int* in_sizes, int n_in,
                              void* d_out, int out_size, void* d_ws, size_t ws_size,
                              hipStream_t stream) {
  (void)in_sizes; (void)n_in; (void)out_size; (void)ws_size;

  const float* x_s = (const float*)d_in[0];
  const int*   eis = (const int*)d_in[1];
  const float* x_t = (const float*)d_in[4];
  const int*   eit = (const int*)d_in[5];

  GnnW pe = {(const float*)d_in[8],  (const float*)d_in[9],  (const float*)d_in[10],
             (const float*)d_in[11], (const float*)d_in[12], (const float*)d_in[13],
             (const float*)d_in[14], (const float*)d_in[15]};
  GnnW pc = {(const float*)d_in[16], (const float*)d_in[17], (const float*)d_in[18],
             (const float*)d_in[19], (const float*)d_in[20], (const float*)d_in[21],
             (const float*)d_in[22], (const float*)d_in[23]};
  const float* m_w1 = (const float*)d_in[24];
  const float* m_b1 = (const float*)d_in[25];
  const float* m_w2 = (const float*)d_in[26];
  const float* m_b2 = (const float*)d_in[27];

  const int* src_s = eis;
  const int* dst_s = eis + NE;
  const int* src_t = eit;
  const int* dst_t = eit + NE;

  // ---- workspace carve-up (floats) ----
  float* ws = (float*)d_ws;
  size_t off = 0;
  auto alloc = [&](size_t nfl) { float* p = ws + off; off += nfl; return p; };
  float* Hs   = alloc((size_t)NN * COUT);
  float* Ht   = alloc((size_t)NN * COUT);
  float* Y1   = alloc((size_t)NN * COUT);
  float* Y2   = alloc((size_t)NN * COUT);
  float* Hb   = alloc((size_t)NN * COUT);
  float* Hn   = alloc((size_t)NN * COUT);
  float* Rs   = alloc((size_t)NN * CIN);
  float* Rt   = alloc((size_t)NN * CIN);
  float* Os   = alloc((size_t)NN * COUT);
  float* Ot   = alloc((size_t)NN * COUT);
  float* Ab   = alloc((size_t)NN * HM);
  float* Bb   = alloc((size_t)NN * HM);
  float* Shat = alloc((size_t)NB * NPER * NPER);
  float* Ssm  = alloc((size_t)NB * NPER * NPER);
  float* degs = alloc(4 * (size_t)NN);
  float* inv_src_s = degs;
  float* inv_dst_s = degs + NN;
  float* inv_src_t = degs + 2 * NN;
  float* inv_dst_t = degs + 3 * NN;

  // ---- degrees (mean denominators); edges fixed across all steps ----
  k_fill0<<<(4 * NN + 255) / 256, 256, 0, stream>>>(degs, 4 * NN);
  k_count2<<<(NE + 255) / 256, 256, 0, stream>>>(src_s, dst_s, inv_src_s, inv_dst_s, NE);
  k_count2<<<(NE + 255) / 256, 256, 0, stream>>>(src_t, dst_t, inv_src_t, inv_dst_t, NE);
  k_invert<<<(4 * NN + 255) / 256, 256, 0, stream>>>(degs, 4 * NN);

  // ---- embedding GNN (pe weights) on both sides ----
  run_gnn(stream, x_s, src_s, dst_s, inv_dst_s, inv_src_s, pe, Hs, Y1, Y2, Hb, Hn);
  run_gnn(stream, x_t, src_t, dst_t, inv_dst_t, inv_src_t, pe, Ht, Y1, Y2, Hb, Hn);

  // ---- S_hat[b] = Hs[b] @ Ht[b]^T  (batched NT GEMM, K=512) ----
  {
    dim3 grid(NPER / 32, NPER / 64, NB);
    k_gemm_wmma<false, true><<<grid, 256, 0, stream>>>(
        Hs, Ht, Shat, COUT, COUT, COUT, NPER,
        (long long)NPER * COUT, (long long)NPER * COUT, (long long)NPER * NPER);
  }

  // ---- S_0 = softmax(S_hat) ----
  float* out0 = (float*)d_out;
  float* outL = (float*)d_out + (size_t)NN * NPER;
  k_softmax128<<<(NN * 32) / 256, 256, 0, stream>>>(Shat, out0, NN);

  // ---- refinement loop ----
  for (int t = 0; t < NSTEPS; ++t) {
    k_softmax128<<<(NN * 32) / 256, 256, 0, stream>>>(Shat, Ssm, NN);
    k_rng_normal<<<(NN * CIN / 2 + 255) / 256, 256, 0, stream>>>(Rs, NN * CIN, (unsigned)t);

    // R_t[b] = S[b]^T @ R_s[b]  (batched TN GEMM, M=128, N=512, K=128)
    {
      dim3 grid(CIN / 32, NPER / 64, NB);
      k_gemm_wmma<true, false><<<grid, 256, 0, stream>>>(
          Ssm, Rs, Rt, NPER, NPER, CIN, CIN,
          (long long)NPER * NPER, (long long)NPER * CIN, (long long)NPER * CIN);
    }

    run_gnn(stream, Rs, src_s, dst_s, inv_dst_s, inv_src_s, pc, Os, Y1, Y2, Hb, Hn);
    run_gnn(stream, Rt, src_t, dst_t, inv_dst_t, inv_src_t, pc, Ot, Y1, Y2, Hb, Hn);

    // A = O_s @ m_w1 + m_b1 ; Bm = O_t @ m_w1
    run_gemm(stream, Os, m_w1, m_b1,    Ab, NN, HM, COUT, COUT, HM, HM, 0);
    run_gemm(stream, Ot, m_w1, nullptr, Bb, NN, HM, COUT, COUT, HM, HM, 0);

    // S_hat += relu(A_i - Bm_j) . w2 + b2
    {
      dim3 grid(NPER / 16, NPER / 16, NB);
      k_pairwise<<<grid, 256, 0, stream>>>(Ab, Bb, m_w2, m_b2, Shat);
    }
  }

  // ---- S_L = softmax(S_hat) ----
  k_softmax128<<<(NN * 32) / 256, 256, 0, stream>>>(Shat, outL, NN);
}